// PaiNNBlock_54400055771905
// MI455X (gfx1250) — compile-verified
//
#include <hip/hip_runtime.h>
#include <hip/hip_bf16.h>

// ---------------- problem constants ----------------
constexpr int NN   = 20000;    // nodes
constexpr int EE   = 320000;   // edges
constexpr int CC   = 128;      // channels
constexpr int BB   = 20;       // basis dim
constexpr float RCUT_F = 5.0f;
constexpr float PI_F   = 3.14159265358979323846f;

typedef __attribute__((ext_vector_type(2))) float v2f;
typedef __attribute__((ext_vector_type(8))) float v8f;

// ---------------- f32 WMMA 16x16x4 ----------------
__device__ __forceinline__ v8f wmma4(v2f a, v2f b, v8f c) {
  // (neg_a, A, neg_b, B, c_mod, C, reuse_a, reuse_b)
  return __builtin_amdgcn_wmma_f32_16x16x4_f32(false, a, false, b, (short)0, c,
                                               false, false);
}

// Accumulate one 16x16 f32 tile: acc += A[16 x K] * B[K x 16].
// aRow: pointer to this lane's A row (row0 + (lane&15)); 8B-aligned, even stride.
// bCol: pointer to this lane's B column (col0 + (lane&15)); LDB = B row stride.
// hi = lane>>4 selects K sub-pair per ISA f32 fragment layout.
template<int K, int LDB>
__device__ __forceinline__ void gemm_k(const float* __restrict__ aRow,
                                       const float* __restrict__ bCol,
                                       int hi, v8f& acc) {
#pragma unroll
  for (int k = 0; k < K; k += 4) {
    const int ka = k + 2 * hi;
    v2f a = *(const v2f*)(aRow + ka);
    v2f b;
    b.x = bCol[ka * LDB];
    b.y = bCol[(ka + 1) * LDB];
    acc = wmma4(a, b, acc);
  }
}

__device__ __forceinline__ float silu_f(float z) {
  return z / (1.0f + __expf(-z));
}

// ---------------- Kernel 1: x = silu(q@W1+b1)@W2+b2  [N,384] ----------------
__global__ __launch_bounds__(256) void ctx_kernel(
    const float* __restrict__ q, const float* __restrict__ W1,
    const float* __restrict__ b1, const float* __restrict__ W2,
    const float* __restrict__ b2, float* __restrict__ x) {
  __shared__ float h[16 * 128];
  const int tile = blockIdx.x;
  const int lane = threadIdx.x & 31;
  const int wave = threadIdx.x >> 5;           // 0..7
  const int l15 = lane & 15, hi = lane >> 4;
  const int row0 = tile * 16;
  const float* aRow = q + (size_t)(row0 + l15) * CC;

  { // stage 1: hidden tile (col0 = wave*16), K = 128
    const int col0 = wave * 16;
    v8f acc = {};
    gemm_k<128, 128>(aRow, W1 + col0 + l15, hi, acc);
    const int c = col0 + l15;
    const float bb = b1[c];
#pragma unroll
    for (int v = 0; v < 8; ++v) {
      const int m = v + 8 * hi;
      h[m * 128 + c] = silu_f(acc[v] + bb);
    }
  }
  __syncthreads();
  // stage 2: 24 col tiles of x, 3 per wave, K = 128 from LDS hidden
  const float* hRow = h + l15 * 128;
#pragma unroll
  for (int t = 0; t < 3; ++t) {
    const int col0 = (wave + t * 8) * 16;
    v8f acc = {};
    gemm_k<128, 384>(hRow, W2 + col0 + l15, hi, acc);
    const int c = col0 + l15;
    const float bb = b2[c];
#pragma unroll
    for (int v = 0; v < 8; ++v) {
      const int m = v + 8 * hi;
      x[(size_t)(row0 + m) * 384 + c] = acc[v] + bb;
    }
  }
}

// ---------------- Kernel 2: init accumulators q1=q, mu1=mu ----------------
__global__ __launch_bounds__(256) void init_kernel(
    const float* __restrict__ q, const float* __restrict__ mu,
    float* __restrict__ q1, float* __restrict__ mu1) {
  const size_t idx = (size_t)blockIdx.x * blockDim.x + threadIdx.x;
  const size_t nq = (size_t)NN * CC;
  const size_t nm = (size_t)NN * 3 * CC;
  if (idx < nq) q1[idx] = q[idx];
  else if (idx < nq + nm) mu1[idx - nq] = mu[idx - nq];
}

// ---------------- Kernel 3: edge filter (WMMA) + gather + scatter ----------
__global__ __launch_bounds__(256) void edge_kernel(
    const float* __restrict__ attrs, const int* __restrict__ eidx,
    const float* __restrict__ ew, const float* __restrict__ vers,
    const float* __restrict__ Wf, const float* __restrict__ bfil,
    const float* __restrict__ x, const float* __restrict__ mu,
    float* __restrict__ q1, float* __restrict__ mu1) {
  __shared__ int   s_i[16], s_j[16];
  __shared__ float s_fc[16];
  __shared__ float s_v[16][3];
  const int e0 = blockIdx.x * 16;
  const int t = threadIdx.x;
  if (t < 16) {
    const int e = e0 + t;
    s_i[t] = eidx[e];          // receivers
    s_j[t] = eidx[EE + e];     // senders
    const float d = ew[e];
    const float fc = 0.5f * (__cosf(PI_F * d / RCUT_F) + 1.0f);
    s_fc[t] = (d < RCUT_F) ? fc : 0.0f;
    s_v[t][0] = vers[e * 3 + 0];
    s_v[t][1] = vers[e * 3 + 1];
    s_v[t][2] = vers[e * 3 + 2];
  }
  __syncthreads();
  const int lane = threadIdx.x & 31, wave = threadIdx.x >> 5;
  const int l15 = lane & 15, hi = lane >> 4;
  const float* aRow = attrs + (size_t)(e0 + l15) * BB;   // 80B stride, 8B aligned

  // --- dq tile: filter cols [wave*16, wave*16+16) of section 0 ---
  {
    const int c = wave * 16 + l15;
    v8f acc = {};
    gemm_k<20, 384>(aRow, Wf + c, hi, acc);
    const float bb = bfil[c];
#pragma unroll
    for (int v = 0; v < 8; ++v) {
      const int m = v + 8 * hi;
      const float w = (acc[v] + bb) * s_fc[m];
      const int j = s_j[m], i = s_i[m];
      const float val = w * x[(size_t)j * 384 + c];
      atomicAdd(&q1[(size_t)i * CC + c], val);
    }
  }
  // --- paired dmuR (section 1) + dmumu (section 2) for same channel c ---
  {
    const int c = wave * 16 + l15;                 // channel in [0,128)
    v8f accR = {}, accM = {};
    gemm_k<20, 384>(aRow, Wf + 128 + c, hi, accR);
    gemm_k<20, 384>(aRow, Wf + 256 + c, hi, accM);
    const float bR = bfil[128 + c], bM = bfil[256 + c];
#pragma unroll
    for (int v = 0; v < 8; ++v) {
      const int m = v + 8 * hi;
      const float fc = s_fc[m];
      const int j = s_j[m], i = s_i[m];
      const float wR = (accR[v] + bR) * fc;
      const float wM = (accM[v] + bM) * fc;
      const float xR = wR * x[(size_t)j * 384 + 128 + c];
      const float xM = wM * x[(size_t)j * 384 + 256 + c];
      const float* muj = mu + (size_t)j * 384 + c;
      float* mui = mu1 + (size_t)i * 384 + c;
#pragma unroll
      for (int vd = 0; vd < 3; ++vd) {
        const float contrib = xR * s_v[m][vd] + xM * muj[vd * 128];
        atomicAdd(&mui[vd * 128], contrib);
      }
    }
  }
}

// ---------------- Kernel 4a: mu_mix = (mu1 as [3N,128]) @ W_mix [128,256] ---
__global__ __launch_bounds__(256) void mumix_kernel(
    const float* __restrict__ mu1, const float* __restrict__ Wmix,
    float* __restrict__ mumix) {
  const int tile = blockIdx.x;                 // 3N/16 tiles
  const int lane = threadIdx.x & 31;
  const int wave = threadIdx.x >> 5;
  const int l15 = lane & 15, hi = lane >> 4;
  const int row0 = tile * 16;
  const float* aRow = mu1 + (size_t)(row0 + l15) * 128;
#pragma unroll
  for (int t = 0; t < 2; ++t) {
    const int col0 = (wave + t * 8) * 16;      // 16 col tiles over 256 cols
    v8f acc = {};
    gemm_k<128, 256>(aRow, Wmix + col0 + l15, hi, acc);
    const int c = col0 + l15;
#pragma unroll
    for (int v = 0; v < 8; ++v) {
      const int m = v + 8 * hi;
      mumix[(size_t)(row0 + m) * 256 + c] = acc[v];
    }
  }
}

// ---------------- Kernel 4b: mu_Vn = sqrt(sum_v V^2+eps), dot = sum_v V*W ---
__global__ __launch_bounds__(256) void vndot_kernel(
    const float* __restrict__ mumix, float* __restrict__ muVn,
    float* __restrict__ dotVW) {
  const int idx = blockIdx.x * blockDim.x + threadIdx.x;
  if (idx >= NN * CC) return;
  const int n = idx >> 7, c = idx & 127;
  const float* base = mumix + (size_t)n * 768;   // [3][256]
  float vsum = 1e-8f, d = 0.0f;
#pragma unroll
  for (int vd = 0; vd < 3; ++vd) {
    const float V = base[vd * 256 + c];
    const float W = base[vd * 256 + 128 + c];
    vsum += V * V;
    d += V * W;
  }
  muVn[idx] = sqrtf(vsum);
  dotVW[idx] = d;
}

// ---------------- Kernel 4c: y = silu([q1,muVn]@Wm1+bm1)@Wm2+bm2 -----------
__global__ __launch_bounds__(256) void mixmlp_kernel(
    const float* __restrict__ q1, const float* __restrict__ muVn,
    const float* __restrict__ Wm1, const float* __restrict__ bm1,
    const float* __restrict__ Wm2, const float* __restrict__ bm2,
    float* __restrict__ y) {
  __shared__ float h[16 * 128];
  const int tile = blockIdx.x;
  const int lane = threadIdx.x & 31;
  const int wave = threadIdx.x >> 5;
  const int l15 = lane & 15, hi = lane >> 4;
  const int row0 = tile * 16;

  { // stage 1: concat-context GEMM, K = 256 split across two A sources
    const int col0 = wave * 16;
    v8f acc = {};
    gemm_k<128, 128>(q1 + (size_t)(row0 + l15) * 128, Wm1 + col0 + l15, hi, acc);
    gemm_k<128, 128>(muVn + (size_t)(row0 + l15) * 128,
                     Wm1 + 128 * 128 + col0 + l15, hi, acc);
    const int c = col0 + l15;
    const float bb = bm1[c];
#pragma unroll
    for (int v = 0; v < 8; ++v) {
      const int m = v + 8 * hi;
      h[m * 128 + c] = silu_f(acc[v] + bb);
    }
  }
  __syncthreads();
  const float* hRow = h + l15 * 128;
#pragma unroll
  for (int t = 0; t < 3; ++t) {
    const int col0 = (wave + t * 8) * 16;
    v8f acc = {};
    gemm_k<128, 384>(hRow, Wm2 + col0 + l15, hi, acc);
    const int c = col0 + l15;
    const float bb = bm2[c];
#pragma unroll
    for (int v = 0; v < 8; ++v) {
      const int m = v + 8 * hi;
      y[(size_t)(row0 + m) * 384 + c] = acc[v] + bb;
    }
  }
}

// ---------------- Kernel 4d: final combine into d_out ----------------------
__global__ __launch_bounds__(256) void finalize_kernel(
    const float* __restrict__ q1, const float* __restrict__ mu1,
    const float* __restrict__ mumix, const float* __restrict__ dotVW,
    const float* __restrict__ y, float* __restrict__ qout,
    float* __restrict__ muout) {
  const int idx = blockIdx.x * blockDim.x + threadIdx.x;
  if (idx >= NN * CC) return;
  const int n = idx >> 7, c = idx & 127;
  const float* yrow = y + (size_t)n * 384;
  const float dq = yrow[c];
  const float dmu = yrow[128 + c];
  const float dqmu = yrow[256 + c];
  qout[idx] = q1[idx] + dq + dqmu * dotVW[idx];
#pragma unroll
  for (int vd = 0; vd < 3; ++vd) {
    const size_t mi = (size_t)(n * 3 + vd) * 128 + c;
    const float muW = mumix[(size_t)n * 768 + vd * 256 + 128 + c];
    muout[mi] = mu1[mi] + dmu * muW;
  }
}

// ---------------- host launcher ----------------
extern "C" void kernel_launch(void* const* d_in, const int* in_sizes, int n_in,
                              void* d_out, int out_size, void* d_ws,
                              size_t ws_size, hipStream_t stream) {
  const float* q    = (const float*)d_in[0];
  const float* mu   = (const float*)d_in[1];
  const int*   eidx = (const int*)  d_in[2];
  const float* ew   = (const float*)d_in[3];
  const float* vers = (const float*)d_in[4];
  const float* attr = (const float*)d_in[5];
  const float* Wf   = (const float*)d_in[6];
  const float* bfil = (const float*)d_in[7];
  const float* W1   = (const float*)d_in[8];
  const float* b1   = (const float*)d_in[9];
  const float* W2   = (const float*)d_in[10];
  const float* b2   = (const float*)d_in[11];
  const float* Wmix = (const float*)d_in[12];
  const float* Wm1  = (const float*)d_in[13];
  const float* bm1  = (const float*)d_in[14];
  const float* Wm2  = (const float*)d_in[15];
  const float* bm2  = (const float*)d_in[16];

  float* ws    = (float*)d_ws;
  float* x     = ws;                           // N*384
  float* q1    = x     + (size_t)NN * 384;     // N*128
  float* mu1   = q1    + (size_t)NN * 128;     // N*384
  float* mumix = mu1   + (size_t)NN * 384;     // N*768  ([3N,256])
  float* muVn  = mumix + (size_t)NN * 768;     // N*128
  float* dotVW = muVn  + (size_t)NN * 128;     // N*128
  float* y     = dotVW + (size_t)NN * 128;     // N*384

  float* qout  = (float*)d_out;                // N*128
  float* muout = qout + (size_t)NN * 128;      // N*384

  ctx_kernel<<<NN / 16, 256, 0, stream>>>(q, W1, b1, W2, b2, x);
  init_kernel<<<(NN * 512 + 255) / 256, 256, 0, stream>>>(q, mu, q1, mu1);
  edge_kernel<<<EE / 16, 256, 0, stream>>>(attr, eidx, ew, vers, Wf, bfil, x,
                                           mu, q1, mu1);
  mumix_kernel<<<(3 * NN) / 16, 256, 0, stream>>>(mu1, Wmix, mumix);
  vndot_kernel<<<(NN * CC + 255) / 256, 256, 0, stream>>>(mumix, muVn, dotVW);
  mixmlp_kernel<<<NN / 16, 256, 0, stream>>>(q1, muVn, Wm1, bm1, Wm2, bm2, y);
  finalize_kernel<<<(NN * CC + 255) / 256, 256, 0, stream>>>(q1, mu1, mumix,
                                                             dotVW, y, qout,
                                                             muout);
}